// GATNet_87857851007401
// MI455X (gfx1250) — compile-verified
//
#include <hip/hip_runtime.h>
#include <math.h>

typedef float v2f __attribute__((ext_vector_type(2)));
typedef float v8f __attribute__((ext_vector_type(8)));

// ---------- helpers: order-preserving float<->uint for atomicMax ----------
__device__ __forceinline__ unsigned enc_f32(float f) {
    unsigned u = __float_as_uint(f);
    return (u & 0x80000000u) ? ~u : (u | 0x80000000u);
}
__device__ __forceinline__ float dec_f32(unsigned x) {
    unsigned u = (x & 0x80000000u) ? (x ^ 0x80000000u) : ~x;
    return __uint_as_float(u);
}
#define ENC_NEG_INF 0x007FFFFFu  // enc_f32(-inf)

// ---------- WMMA fp32 GEMM: z[n,dout] = x[n,128] @ W[dout,128]^T ----------
// One wave handles a 16-wide output-column tile (j0 = wave*16) and MT row
// tiles of 16. B panel (16 cols x K=128 of W) is preloaded into 32 v2f regs.
// Layouts per CDNA5 ISA 7.12.2:
//   A 16x4 (f32):  lane L, vgpr j -> A[M=L&15,      K=2*(L>>4)+j]
//   B 4x16 (f32):  lane L, vgpr j -> B[K=2*(L>>4)+j, N=L&15]
//   C/D 16x16:     lane L, vgpr v -> C[M=v+8*(L>>4), N=L&15]
template <int MT>
__global__ void gemm_wmma128(const float* __restrict__ x,
                             const float* __restrict__ W,
                             float* __restrict__ z, int n, int dout) {
    const int DIN   = 128;
    const int lane  = threadIdx.x & 31;
    const int wave  = threadIdx.x >> 5;
    const int j0    = wave * 16;          // output column tile base
    const int m     = lane & 15;
    const int khalf = (lane >> 4) * 2;    // 0 or 2

    // Preload B panel: b[s] = W[(j0+m), 4*s + khalf + {0,1}]
    v2f b[DIN / 4];
    const float* wrow = W + (size_t)(j0 + m) * DIN + khalf;
#pragma unroll
    for (int s = 0; s < DIN / 4; ++s)
        b[s] = *(const v2f*)(wrow + 4 * s);

    for (int t = 0; t < MT; ++t) {
        int i0 = (blockIdx.x * MT + t) * 16;
        if (i0 >= n) break;
        int row = i0 + m;
        if (row >= n) row = n - 1;        // clamp; store is masked per-row
        const float* xrow = x + (size_t)row * DIN + khalf;

        v8f acc = {};
#pragma unroll
        for (int s = 0; s < DIN / 4; ++s) {
            v2f a = *(const v2f*)(xrow + 4 * s);
            acc = __builtin_amdgcn_wmma_f32_16x16x4_f32(
                false, a, false, b[s], (short)0, acc, false, false);
        }

        int rbase = i0 + ((lane >> 4) << 3);
#pragma unroll
        for (int v = 0; v < 8; ++v) {
            int r = rbase + v;
            if (r < n) z[(size_t)r * dout + j0 + m] = acc[v];
        }
    }
}

// ---------- per-node attention scalars: es = z.a[:d], ed = z.a[d:] ----------
__global__ void node_scores(const float* __restrict__ z,
                            const float* __restrict__ a,
                            float* __restrict__ es, float* __restrict__ ed,
                            int n, int d) {
    int wid  = (blockIdx.x * blockDim.x + threadIdx.x) >> 5;
    int lane = threadIdx.x & 31;
    if (wid >= n) return;
    const float* zr = z + (size_t)wid * d;
    float s1 = 0.f, s2 = 0.f;
    for (int c = lane; c < d; c += 32) {
        float zv = zr[c];
        s1 += zv * a[c];
        s2 += zv * a[d + c];
    }
#pragma unroll
    for (int off = 16; off; off >>= 1) {
        s1 += __shfl_xor(s1, off, 32);
        s2 += __shfl_xor(s2, off, 32);
    }
    if (lane == 0) { es[wid] = s1; ed[wid] = s2; }
}

// ---------- zero output / init softmax accumulators ----------
__global__ void init_layer(float* __restrict__ out, unsigned* __restrict__ menc,
                           float* __restrict__ ssum, int n, int d) {
    size_t i     = (size_t)blockIdx.x * blockDim.x + threadIdx.x;
    size_t total = (size_t)n * d;
    if (i < total) out[i] = 0.f;
    if (i < (size_t)n) { menc[i] = ENC_NEG_INF; ssum[i] = 0.f; }
}

// ---------- edge pass 1: leaky-relu score * mask, segment max ----------
__global__ void edge_score_max(const float* __restrict__ es,
                               const float* __restrict__ ed,
                               const float* __restrict__ tm,
                               const float* __restrict__ fm,
                               const int* __restrict__ src,
                               const int* __restrict__ dst,
                               float* __restrict__ e_out,
                               unsigned* __restrict__ menc, int E) {
    int i = blockIdx.x * blockDim.x + threadIdx.x;
    if (i >= E) return;
    float v = es[src[i]] + ed[dst[i]];
    v = v > 0.f ? v : 0.01f * v;     // leaky_relu, slope 0.01
    v *= tm[i] * fm[i];
    e_out[i] = v;
    atomicMax(&menc[dst[i]], enc_f32(v));
}

// ---------- edge pass 2: ex = exp(e - m[dst]), segment sum ----------
__global__ void edge_exp_sum(float* __restrict__ e_io,
                             const unsigned* __restrict__ menc,
                             const int* __restrict__ dst,
                             float* __restrict__ ssum, int E) {
    int i = blockIdx.x * blockDim.x + threadIdx.x;
    if (i >= E) return;
    int dn   = dst[i];
    float ex = expf(e_io[i] - dec_f32(menc[dn]));
    e_io[i]  = ex;
    unsafeAtomicAdd(&ssum[dn], ex);
}

// ---------- edge pass 3: out[dst] += (ex/s[dst]) * z[src] (wave/edge) ----------
__global__ void edge_aggregate(const float* __restrict__ ex,
                               const float* __restrict__ ssum,
                               const int* __restrict__ src,
                               const int* __restrict__ dst,
                               const float* __restrict__ z,
                               float* __restrict__ out, int E, int d) {
    int wid  = (blockIdx.x * blockDim.x + threadIdx.x) >> 5;
    int lane = threadIdx.x & 31;
    int nw   = (gridDim.x * blockDim.x) >> 5;
    for (int e = wid; e < E; e += nw) {
        int sn = src[e], dn = dst[e];
        float alpha    = ex[e] / ssum[dn];
        const float* zr = z + (size_t)sn * d;
        float* orow     = out + (size_t)dn * d;
        for (int c = lane; c < d; c += 32)
            unsafeAtomicAdd(&orow[c], alpha * zr[c]);
    }
}

// ============================ host launcher ============================
extern "C" void kernel_launch(void* const* d_in, const int* in_sizes, int n_in,
                              void* d_out, int out_size, void* d_ws, size_t ws_size,
                              hipStream_t stream) {
    const float* h   = (const float*)d_in[0];
    const float* W1  = (const float*)d_in[3];
    const float* a1  = (const float*)d_in[4];
    const float* W2  = (const float*)d_in[5];
    const float* a2  = (const float*)d_in[6];
    const float* tm  = (const float*)d_in[7];
    const float* fm  = (const float*)d_in[8];
    const int*   src = (const int*)d_in[9];
    const int*   dst = (const int*)d_in[10];

    const int N   = in_sizes[1];          // snorm_n is [N,1]
    const int E   = in_sizes[9];          // src is [E]
    const int HID = 128, OUT = 64;

    // Workspace layout (floats): z(N*128) | h1(N*128) | es(N) | ed(N) | menc(N) | s(N) | e(E)
    float*    z     = (float*)d_ws;
    float*    h1    = z + (size_t)N * HID;
    float*    es    = h1 + (size_t)N * HID;
    float*    ed    = es + N;
    unsigned* menc  = (unsigned*)(ed + N);
    float*    ssum  = (float*)(menc + N);
    float*    eedge = ssum + N;
    float*    out   = (float*)d_out;

    const int ROWS_BLK = 64;              // MT=4 row tiles of 16
    dim3 gGemm((N + ROWS_BLK - 1) / ROWS_BLK);
    int  gNode = (N + 7) / 8;             // 8 waves / 256-thread block
    int  gEdge = (E + 255) / 256;

    // ---------------- Layer 1: [N,128] -> [N,128] ----------------
    gemm_wmma128<4><<<gGemm, 8 * 32, 0, stream>>>(h, W1, z, N, HID);
    node_scores<<<gNode, 256, 0, stream>>>(z, a1, es, ed, N, HID);
    init_layer<<<(int)(((size_t)N * HID + 255) / 256), 256, 0, stream>>>(h1, menc, ssum, N, HID);
    edge_score_max<<<gEdge, 256, 0, stream>>>(es, ed, tm, fm, src, dst, eedge, menc, E);
    edge_exp_sum<<<gEdge, 256, 0, stream>>>(eedge, menc, dst, ssum, E);
    edge_aggregate<<<2048, 256, 0, stream>>>(eedge, ssum, src, dst, z, h1, E, HID);

    // ---------------- Layer 2: [N,128] -> [N,64] ----------------
    gemm_wmma128<4><<<gGemm, 4 * 32, 0, stream>>>(h1, W2, z, N, OUT);
    node_scores<<<gNode, 256, 0, stream>>>(z, a2, es, ed, N, OUT);
    init_layer<<<(int)(((size_t)N * OUT + 255) / 256), 256, 0, stream>>>(out, menc, ssum, N, OUT);
    edge_score_max<<<gEdge, 256, 0, stream>>>(es, ed, tm, fm, src, dst, eedge, menc, E);
    edge_exp_sum<<<gEdge, 256, 0, stream>>>(eedge, menc, dst, ssum, E);
    edge_aggregate<<<2048, 256, 0, stream>>>(eedge, ssum, src, dst, z, out, E, OUT);
}